// APPM_13460427505999
// MI455X (gfx1250) — compile-verified
//
#include <hip/hip_runtime.h>
#include <hip/hip_bf16.h>
#include <math.h>

// ---------------------------------------------------------------------------
// Problem constants (match reference)
// ---------------------------------------------------------------------------
#define BATCH   64
#define CHANS   2048
#define FM      14
#define NPOS    (FM * FM)          // 196
#define POS_TILES 13               // ceil(196 / 16)
#define NWIN    361                // 121 + 120 + 120
#define NPICK   7                  // 2 + 3 + 2

typedef float v2f __attribute__((ext_vector_type(2)));
typedef float v8f __attribute__((ext_vector_type(8)));

// ---------------------------------------------------------------------------
// Kernel 1: channel-sum via V_WMMA_F32_16X16X4_F32 with an all-ones A matrix.
//   s[b][pos] = sum_c x[b][c][pos]
// One block = (pos_tile, batch); 8 waves each reduce a 256-channel slab over
// 16 spatial positions in the WMMA fp32 accumulator; LDS combines the waves.
// With A == ones, D[0][n] = sum over K of B[k][n] for ANY (lane,vgpr)->K
// mapping, so we only rely on column index == lane%16 of the B operand.
// Loop is a compile-time 32-trip pointer walk (uniform -> scalar branch),
// with two independent accumulators to break the WMMA RAW chain.
// ---------------------------------------------------------------------------
__global__ __launch_bounds__(256)
void appm_channel_sum_wmma(const float* __restrict__ x, float* __restrict__ s) {
    const int tile = blockIdx.x;        // 0..12  (16 positions each)
    const int b    = blockIdx.y;        // 0..63
    const int lane = threadIdx.x & 31;
    const int wave = threadIdx.x >> 5;  // 0..7

    const int  pos   = tile * 16 + (lane & 15);
    const bool inb   = (pos < NPOS);
    const int  posc  = inb ? pos : 0;                  // safe clamp
    const int  chalf = lane >> 4;                      // 0 or 1

    // Base pointer for this lane: x[b][wave*256 + chalf*2][posc].
    const float* __restrict__ p =
        x + (long)b * CHANS * NPOS
          + (long)(wave * (CHANS / 8) + chalf * 2) * NPOS
          + posc;

    v2f a; a.x = 1.0f; a.y = 1.0f;                     // all-ones A (16x4)
    v8f acc0 = {0.f,0.f,0.f,0.f,0.f,0.f,0.f,0.f};
    v8f acc1 = {0.f,0.f,0.f,0.f,0.f,0.f,0.f,0.f};

    // 256 channels per wave, 4 per WMMA, 2 WMMAs per trip -> 32 trips.
    #pragma unroll 4
    for (int k = 0; k < 32; ++k) {
        float e0 = p[0];
        float e1 = p[NPOS];
        float e2 = p[4 * NPOS];
        float e3 = p[5 * NPOS];
        p += 8 * NPOS;
        e0 = inb ? e0 : 0.0f;                          // cndmask, no divergence
        e1 = inb ? e1 : 0.0f;
        e2 = inb ? e2 : 0.0f;
        e3 = inb ? e3 : 0.0f;
        v2f b0; b0.x = e0; b0.y = e1;
        v2f b1; b1.x = e2; b1.y = e3;
        // 8 args: (neg_a, A, neg_b, B, c_mod, C, reuse_a, reuse_b)
        acc0 = __builtin_amdgcn_wmma_f32_16x16x4_f32(
            false, a, false, b0, (short)0, acc0, false, false);
        acc1 = __builtin_amdgcn_wmma_f32_16x16x4_f32(
            false, a, false, b1, (short)0, acc1, false, false);
    }

    // Row 0 of D lives in element 0 on lanes 0..15 (N = lane).
    const float r = acc0[0] + acc1[0];

    __shared__ float red[8][16];
    if (lane < 16) red[wave][lane] = r;
    __syncthreads();

    if (threadIdx.x < 16) {
        float t = 0.0f;
        #pragma unroll
        for (int w = 0; w < 8; ++w) t += red[w][threadIdx.x];
        const int pp = tile * 16 + threadIdx.x;
        if (pp < NPOS) s[b * NPOS + pp] = t;
    }
}

// ---------------------------------------------------------------------------
// Kernel 2: per-batch window pooling + greedy NMS (tiny; 64 blocks).
// Output layout (all float, reference tuple order, flat):
//   [0,              448)  proposalN_indices  (cast to float)
//   [448,            896)  proposalN_windows_scores
//   [896,          24000)  all_scores [64][361]
// ---------------------------------------------------------------------------
__global__ __launch_bounds__(128)
void appm_pool_nms(const float* __restrict__ s, const int* __restrict__ coords,
                   float* __restrict__ out) {
    const int b = blockIdx.x;

    __shared__ float sm[NPOS];
    __shared__ float sc[NWIN];
    __shared__ unsigned char alive[121];
    __shared__ int picks[NPICK];

    for (int i = threadIdx.x; i < NPOS; i += blockDim.x) sm[i] = s[b * NPOS + i];
    __syncthreads();

    // windows: g0 rh=4 rw=4 (11x11=121, off 0); g1 rh=3 rw=5 (12x10=120, off 121);
    //          g2 rh=5 rw=3 (10x12=120, off 241)
    for (int w = threadIdx.x; w < NWIN; w += blockDim.x) {
        int rh, rw, nc, off;
        if (w < 121)       { off = 0;   rh = 4; rw = 4; nc = 11; }
        else if (w < 241)  { off = 121; rh = 3; rw = 5; nc = 10; }
        else               { off = 241; rh = 5; rw = 3; nc = 12; }
        const int loc = w - off;
        const int i = loc / nc;
        const int j = loc % nc;
        float t = 0.0f;
        for (int di = 0; di < rh; ++di)
            for (int dj = 0; dj < rw; ++dj)
                t += sm[(i + di) * FM + (j + dj)];
        sc[w] = t / (float)(rh * rw);
    }
    __syncthreads();

    for (int w = threadIdx.x; w < NWIN; w += blockDim.x)
        out[2 * BATCH * NPICK + b * NWIN + w] = sc[w];

    if (threadIdx.x == 0) {
        const int starts[4] = {0, 121, 241, 361};
        const int ns[3]     = {2, 3, 2};
        int pi = 0;
        for (int g = 0; g < 3; ++g) {
            const int st = starts[g];
            const int W  = starts[g + 1] - st;
            for (int i = 0; i < W; ++i) alive[i] = 1;
            for (int it = 0; it < ns[g]; ++it) {
                // argmax over alive (first index wins ties; all-dead -> 0,
                // matching jnp.argmax over -inf)
                float best = -INFINITY; int bi = 0;
                for (int i = 0; i < W; ++i) {
                    if (alive[i] && sc[st + i] > best) { best = sc[st + i]; bi = i; }
                }
                picks[pi++] = st + bi;
                const float bx0 = (float)coords[(st + bi) * 4 + 0];
                const float by0 = (float)coords[(st + bi) * 4 + 1];
                const float bx1 = (float)coords[(st + bi) * 4 + 2];
                const float by1 = (float)coords[(st + bi) * 4 + 3];
                const float ba  = (bx1 - bx0) * (by1 - by0);
                for (int i = 0; i < W; ++i) {
                    if (!alive[i]) continue;
                    const float cx0 = (float)coords[(st + i) * 4 + 0];
                    const float cy0 = (float)coords[(st + i) * 4 + 1];
                    const float cx1 = (float)coords[(st + i) * 4 + 2];
                    const float cy1 = (float)coords[(st + i) * 4 + 3];
                    const float ix0 = fmaxf(bx0, cx0), iy0 = fmaxf(by0, cy0);
                    const float ix1 = fminf(bx1, cx1), iy1 = fminf(by1, cy1);
                    const float inter = fmaxf(ix1 - ix0, 0.0f) * fmaxf(iy1 - iy0, 0.0f);
                    const float ca = (cx1 - cx0) * (cy1 - cy0);
                    const float iou = inter / (ba + ca - inter);
                    if (iou >= 0.25f) alive[i] = 0;   // alive &= (iou < thresh)
                }
            }
        }
        for (int k = 0; k < NPICK; ++k) {
            out[b * NPICK + k]                 = (float)picks[k];
            out[BATCH * NPICK + b * NPICK + k] = sc[picks[k]];
        }
    }
}

// ---------------------------------------------------------------------------
extern "C" void kernel_launch(void* const* d_in, const int* in_sizes, int n_in,
                              void* d_out, int out_size, void* d_ws, size_t ws_size,
                              hipStream_t stream) {
    const float* x      = (const float*)d_in[0];   // [64,2048,14,14] fp32
    const int*   coords = (const int*)d_in[1];     // [361,4] int32
    float*       out    = (float*)d_out;           // 24000 floats
    float*       s      = (float*)d_ws;            // [64,196] channel sums

    appm_channel_sum_wmma<<<dim3(POS_TILES, BATCH), 256, 0, stream>>>(x, s);
    appm_pool_nms<<<dim3(BATCH), 128, 0, stream>>>(s, coords, out);
}